// GCN_GNN_70411693851060
// MI455X (gfx1250) — compile-verified
//
#include <hip/hip_runtime.h>

typedef __attribute__((ext_vector_type(2))) float v2f;
typedef __attribute__((ext_vector_type(8))) float v8f;

#define NDIM 128              // feature / hidden width (K == N == 128)
#define ROWS_PER_WAVE 16
#define WAVES_PER_BLOCK 4
#define ROWS_PER_BLOCK (ROWS_PER_WAVE * WAVES_PER_BLOCK)

// ---------------------------------------------------------------------------
// Small utility kernels
// ---------------------------------------------------------------------------
__global__ void fill_kernel(float* __restrict__ p, float v, int n) {
    int i = blockIdx.x * blockDim.x + threadIdx.x;
    if (i < n) p[i] = v;
}

__global__ void rsqrt_kernel(float* __restrict__ p, int n) {
    int i = blockIdx.x * blockDim.x + threadIdx.x;
    if (i < n) p[i] = rsqrtf(p[i]);
}

__global__ void relu_kernel(float* __restrict__ p, int n) {
    int i = blockIdx.x * blockDim.x + threadIdx.x;
    if (i < n) p[i] = fmaxf(p[i], 0.0f);
}

__global__ void deg_scatter_kernel(const int* __restrict__ dst,
                                   float* __restrict__ deg, int e) {
    int i = blockIdx.x * blockDim.x + threadIdx.x;
    if (i < e) atomicAdd(&deg[dst[i]], 1.0f);
}

// out[dst] += x[src]   (one block per edge, one lane per feature)
__global__ void scatter_sum_kernel(const int* __restrict__ src,
                                   const int* __restrict__ dst,
                                   const float* __restrict__ x,
                                   float* __restrict__ out) {
    int e = blockIdx.x;
    int f = threadIdx.x;
    int s = src[e];
    int d = dst[e];
    atomicAdd(&out[(size_t)d * NDIM + f], x[(size_t)s * NDIM + f]);
}

// out[dst] += t[src] * dinv[src] * dinv[dst]
__global__ void scatter_norm_kernel(const int* __restrict__ src,
                                    const int* __restrict__ dst,
                                    const float* __restrict__ t,
                                    const float* __restrict__ dinv,
                                    float* __restrict__ out) {
    int e = blockIdx.x;
    int f = threadIdx.x;
    int s = src[e];
    int d = dst[e];
    float nrm = dinv[s] * dinv[d];
    atomicAdd(&out[(size_t)d * NDIM + f], t[(size_t)s * NDIM + f] * nrm);
}

// out[i] = t[i] * dinv[i]^2 + bias   (GCN self-loop term; also initializes out)
__global__ void selfloop_bias_kernel(const float* __restrict__ t,
                                     const float* __restrict__ dinv,
                                     const float* __restrict__ bias,
                                     float* __restrict__ out) {
    int i = blockIdx.x;
    int f = threadIdx.x;
    float dv = dinv[i];
    out[(size_t)i * NDIM + f] = t[(size_t)i * NDIM + f] * dv * dv + bias[f];
}

__global__ void pool_kernel(const float* __restrict__ h,
                            const int* __restrict__ batch,
                            float* __restrict__ sums,
                            float* __restrict__ counts) {
    int i = blockIdx.x;
    int f = threadIdx.x;
    int g = batch[i];
    atomicAdd(&sums[(size_t)g * NDIM + f], h[(size_t)i * NDIM + f]);
    if (f == 0) atomicAdd(&counts[g], 1.0f);
}

__global__ void embed_kernel(const float* __restrict__ sums,
                             const float* __restrict__ counts,
                             float* __restrict__ emb) {
    int g = blockIdx.x;
    int f = threadIdx.x;
    emb[(size_t)g * NDIM + f] = sums[(size_t)g * NDIM + f] / fmaxf(counts[g], 1.0f);
}

// logits[g,c] = emb[g,:] . lin_w[:,c] + lin_b[c]   (tiny: one thread per output)
__global__ void logits_kernel(const float* __restrict__ emb,
                              const float* __restrict__ lw,
                              const float* __restrict__ lb,
                              float* __restrict__ out, int C) {
    int t = threadIdx.x;
    int g = t / C;
    int c = t % C;
    float acc = lb[c];
    for (int k = 0; k < NDIM; ++k)
        acc += emb[(size_t)g * NDIM + k] * lw[k * C + c];
    out[g * C + c] = acc;
}

// ---------------------------------------------------------------------------
// WMMA GEMM: out[nrows,128] = A0 @ W0 (+ A1 @ W1) (+ bias) (relu?)
// Block = 128 threads (4 wave32). Each wave computes 16 rows x 128 cols via
// 8 accumulator tiles and V_WMMA_F32_16X16X4_F32 along K (steps of 4).
//
// The weight matrix is staged in LDS *pre-packed as K-pairs*:
//   ldsW[(k>>1)*NDIM + c] = { W[k][c], W[k+1][c] }   (k even)
// so each lane's B operand is one aligned ds_load_b64 directly into an even
// VGPR pair (no repacking movs before the WMMA). Total LDS = 64 KB.
// ---------------------------------------------------------------------------
__global__ void __launch_bounds__(128)
wmma_gemm128(const float* __restrict__ A0, const float* __restrict__ W0,
             const float* __restrict__ A1, const float* __restrict__ W1,
             const float* __restrict__ bias, float* __restrict__ out,
             int nrows, int do_relu) {
    __shared__ v2f ldsW[(NDIM / 2) * NDIM];      // 64 * 128 * 8B = 64 KB

    const int tid  = threadIdx.x;
    const int wave = tid >> 5;
    const int lane = tid & 31;
    const int half = lane >> 4;                  // 0: K pair {k0,k0+1}; 1: {k0+2,k0+3}
    const int r    = lane & 15;                  // row (A) / column (B,C,D)
    const int row0 = blockIdx.x * ROWS_PER_BLOCK + wave * ROWS_PER_WAVE;

    // init accumulators with bias (broadcast down rows) or zero
    v8f acc[8];
    for (int nt = 0; nt < 8; ++nt) {
        float bv = (bias != nullptr) ? bias[nt * 16 + r] : 0.0f;
        for (int v = 0; v < 8; ++v) acc[nt][v] = bv;
    }

    // clamp A row so tail lanes stay in-bounds (EXEC stays all-ones)
    int arow = row0 + r;
    if (arow >= nrows) arow = nrows - 1;

    // ---- pass 0: A0 @ W0 ----
    for (int i = tid; i < (NDIM / 2) * NDIM; i += blockDim.x) {
        int kp = i >> 7;                         // i / NDIM
        int c  = i & (NDIM - 1);                 // i % NDIM
        v2f w;
        w.x = W0[(2 * kp + 0) * NDIM + c];
        w.y = W0[(2 * kp + 1) * NDIM + c];
        ldsW[i] = w;
    }
    __syncthreads();
    for (int k0 = 0; k0 < NDIM; k0 += 4) {
        const v2f a = *(const v2f*)(A0 + (size_t)arow * NDIM + k0 + 2 * half);
        const int kp = (k0 >> 1) + half;
        for (int nt = 0; nt < 8; ++nt) {
            const v2f b = ldsW[kp * NDIM + nt * 16 + r];
            acc[nt] = __builtin_amdgcn_wmma_f32_16x16x4_f32(
                false, a, false, b, (short)0, acc[nt], false, false);
        }
    }

    // ---- pass 1 (GraphConv fusion): + A1 @ W1 ----
    if (A1 != nullptr) {
        __syncthreads();
        for (int i = tid; i < (NDIM / 2) * NDIM; i += blockDim.x) {
            int kp = i >> 7;
            int c  = i & (NDIM - 1);
            v2f w;
            w.x = W1[(2 * kp + 0) * NDIM + c];
            w.y = W1[(2 * kp + 1) * NDIM + c];
            ldsW[i] = w;
        }
        __syncthreads();
        for (int k0 = 0; k0 < NDIM; k0 += 4) {
            const v2f a = *(const v2f*)(A1 + (size_t)arow * NDIM + k0 + 2 * half);
            const int kp = (k0 >> 1) + half;
            for (int nt = 0; nt < 8; ++nt) {
                const v2f b = ldsW[kp * NDIM + nt * 16 + r];
                acc[nt] = __builtin_amdgcn_wmma_f32_16x16x4_f32(
                    false, a, false, b, (short)0, acc[nt], false, false);
            }
        }
    }

    // ---- store: C/D layout -> row = row0 + v + 8*half, col = nt*16 + r ----
    for (int nt = 0; nt < 8; ++nt) {
        for (int v = 0; v < 8; ++v) {
            int row = row0 + v + 8 * half;
            if (row < nrows) {
                float val = acc[nt][v];
                if (do_relu) val = fmaxf(val, 0.0f);
                out[(size_t)row * NDIM + nt * 16 + r] = val;
            }
        }
    }
}

// ---------------------------------------------------------------------------
extern "C" void kernel_launch(void* const* d_in, const int* in_sizes, int n_in,
                              void* d_out, int out_size, void* d_ws, size_t ws_size,
                              hipStream_t stream) {
    const float* x       = (const float*)d_in[0];
    const int*   ei      = (const int*)  d_in[1];
    const int*   batch   = (const int*)  d_in[2];
    const float* w1_root = (const float*)d_in[3];
    const float* w1_rel  = (const float*)d_in[4];
    const float* b1      = (const float*)d_in[5];
    const float* w2_root = (const float*)d_in[6];
    const float* w2_rel  = (const float*)d_in[7];
    const float* b2      = (const float*)d_in[8];
    const float* w3      = (const float*)d_in[9];
    const float* b3      = (const float*)d_in[10];
    const float* w4      = (const float*)d_in[11];
    const float* b4      = (const float*)d_in[12];
    const float* w5      = (const float*)d_in[13];
    const float* b5      = (const float*)d_in[14];
    const float* lin_w   = (const float*)d_in[15];
    const float* lin_b   = (const float*)d_in[16];

    const int N = in_sizes[0] / NDIM;          // 40000
    const int E = in_sizes[1] / 2;             // 600000
    const int C = in_sizes[16];                // 6
    const int G = out_size / (NDIM + C);       // 32

    const int* src = ei;
    const int* dst = ei + E;

    float* ws     = (float*)d_ws;
    const size_t NH = (size_t)N * NDIM;
    float* bufA   = ws;                        // N*H
    float* bufB   = ws + NH;                   // N*H
    float* bufT   = ws + 2 * NH;               // N*H
    float* dinv   = ws + 3 * NH;               // N   (deg -> deg^-1/2 in place)
    float* sums   = dinv + N;                  // G*H
    float* counts = sums + (size_t)G * NDIM;   // G

    const int NHi     = (int)NH;
    const int gemmGrd = (N + ROWS_PER_BLOCK - 1) / ROWS_PER_BLOCK;

    // ---- degree (with self-loops) -> deg^{-1/2} ----
    fill_kernel<<<(N + 255) / 256, 256, 0, stream>>>(dinv, 1.0f, N);
    deg_scatter_kernel<<<(E + 255) / 256, 256, 0, stream>>>(dst, dinv, E);
    rsqrt_kernel<<<(N + 255) / 256, 256, 0, stream>>>(dinv, N);

    // ---- layer 1: GraphConv(x) -> bufA ----
    fill_kernel<<<(NHi + 255) / 256, 256, 0, stream>>>(bufT, 0.0f, NHi);
    scatter_sum_kernel<<<E, NDIM, 0, stream>>>(src, dst, x, bufT);
    wmma_gemm128<<<gemmGrd, 128, 0, stream>>>(x, w1_root, bufT, w1_rel, b1,
                                              bufA, N, 1);

    // ---- layer 2: GraphConv(bufA) -> bufB ----
    fill_kernel<<<(NHi + 255) / 256, 256, 0, stream>>>(bufT, 0.0f, NHi);
    scatter_sum_kernel<<<E, NDIM, 0, stream>>>(src, dst, bufA, bufT);
    wmma_gemm128<<<gemmGrd, 128, 0, stream>>>(bufA, w2_root, bufT, w2_rel, b2,
                                              bufB, N, 1);

    // ---- layer 3: GCNConv(bufB) -> bufA, relu ----
    wmma_gemm128<<<gemmGrd, 128, 0, stream>>>(bufB, w3, nullptr, nullptr,
                                              nullptr, bufT, N, 0);
    selfloop_bias_kernel<<<N, NDIM, 0, stream>>>(bufT, dinv, b3, bufA);
    scatter_norm_kernel<<<E, NDIM, 0, stream>>>(src, dst, bufT, dinv, bufA);
    relu_kernel<<<(NHi + 255) / 256, 256, 0, stream>>>(bufA, NHi);

    // ---- layer 4: GCNConv(bufA) -> bufB, relu ----
    wmma_gemm128<<<gemmGrd, 128, 0, stream>>>(bufA, w4, nullptr, nullptr,
                                              nullptr, bufT, N, 0);
    selfloop_bias_kernel<<<N, NDIM, 0, stream>>>(bufT, dinv, b4, bufB);
    scatter_norm_kernel<<<E, NDIM, 0, stream>>>(src, dst, bufT, dinv, bufB);
    relu_kernel<<<(NHi + 255) / 256, 256, 0, stream>>>(bufB, NHi);

    // ---- layer 5: GCNConv(bufB) -> bufA (no relu) ----
    wmma_gemm128<<<gemmGrd, 128, 0, stream>>>(bufB, w5, nullptr, nullptr,
                                              nullptr, bufT, N, 0);
    selfloop_bias_kernel<<<N, NDIM, 0, stream>>>(bufT, dinv, b5, bufA);
    scatter_norm_kernel<<<E, NDIM, 0, stream>>>(src, dst, bufT, dinv, bufA);

    // ---- global mean pool + head ----
    const int poolN = G * NDIM + G;            // sums + counts are contiguous
    fill_kernel<<<(poolN + 255) / 256, 256, 0, stream>>>(sums, 0.0f, poolN);
    pool_kernel<<<N, NDIM, 0, stream>>>(bufA, batch, sums, counts);

    float* out_logits = (float*)d_out;         // [G, C]
    float* out_emb    = out_logits + G * C;    // [G, H]
    embed_kernel<<<G, NDIM, 0, stream>>>(sums, counts, out_emb);
    logits_kernel<<<1, G * C, 0, stream>>>(out_emb, lin_w, lin_b, out_logits, C);
}